// ModelNew_63806034150027
// MI455X (gfx1250) — compile-verified
//
#include <hip/hip_runtime.h>

// Fused ConvTranspose3d(32->64,k3,s2,p1) + scale + 2x2x2 avgpool + bias + scale,
// algebraically collapsed to a dense 2x2x2 stride-1 conv with effective weights,
// computed as implicit GEMM with V_WMMA_F32_16X16X4_F32 (full fp32 precision).

typedef float v2f __attribute__((ext_vector_type(2)));
typedef float v4f __attribute__((ext_vector_type(4)));
typedef float v8f __attribute__((ext_vector_type(8)));

// ---------------------------------------------------------------------------
// Prep: Weff'[t][ci][co] = scale1*scale2/8 * sum of kernel taps;  t = a*4+b*2+c
//       (a=0 -> kd in {1,2}; a=1 -> kd in {0}; same per dim)
//       beta[co] = scale2*(scale1*conv_bias[co] + bias[co])
// ---------------------------------------------------------------------------
__global__ __launch_bounds__(256) void ModelNew_prep_kernel(
    const float* __restrict__ w, const float* __restrict__ conv_bias,
    const float* __restrict__ bias, const float* __restrict__ s1p,
    const float* __restrict__ s2p, float* __restrict__ weff,
    float* __restrict__ beta)
{
    const int idx = blockIdx.x * 256 + threadIdx.x;
    const float s1 = s1p[0], s2 = s2p[0];
    if (idx < 8 * 32 * 64) {
        const int co = idx & 63;
        const int ci = (idx >> 6) & 31;
        const int t  = idx >> 11;
        const int a = (t >> 2) & 1, b = (t >> 1) & 1, c = t & 1;
        const float* wp = w + (ci * 64 + co) * 27;
        float sum = 0.f;
        for (int kd = 0; kd < 3; ++kd) {
            if (a ? (kd != 0) : (kd == 0)) continue;
            for (int kh = 0; kh < 3; ++kh) {
                if (b ? (kh != 0) : (kh == 0)) continue;
                for (int kw = 0; kw < 3; ++kw) {
                    if (c ? (kw != 0) : (kw == 0)) continue;
                    sum += wp[kd * 9 + kh * 3 + kw];
                }
            }
        }
        weff[idx] = sum * (s1 * s2 * 0.125f);
    }
    if (idx < 64) {
        beta[idx] = s2 * (s1 * conv_bias[idx] + bias[idx]);
    }
}

// ---------------------------------------------------------------------------
// Main: one block per (n, do, ho). 8 waves = (2 wo-tiles) x (4 co-tiles).
// Each wave: 16(wo) x 16(co) output tile, K = 8 taps * 32 ci via 64 WMMAs.
// Both A and B operands are ci-innermost in LDS so every WMMA operand pair is
// one aligned ds_load_b64 (fusable to ds_load_2addr_b64), no packing movs.
// ---------------------------------------------------------------------------
__global__ __launch_bounds__(256) void ModelNew_fused_kernel(
    const float* __restrict__ x, const float* __restrict__ weff,
    const float* __restrict__ beta, float* __restrict__ out)
{
    // Weff' staged as [t][co][ci], row stride 34 floats (even -> pairs 8B-aligned;
    // 34*l mod 64 distinct across lanes -> conflict-free b64 reads)
    __shared__ float ldsB[8 * 64 * 34];
    // x tile as [q = z*2+y][col(32)][ci], column stride 34 floats: A operand pair
    // (ci, ci+1) is contiguous + 8B aligned; 34-stride de-conflicts the 16 cols.
    __shared__ float ldsX[4 * 32 * 34];

    const int tid = threadIdx.x;

    // ---- stage Weff' ----
    #pragma unroll 4
    for (int i = 0; i < 64; ++i) {
        const int idx = tid + i * 256;          // (t*32+ci)*64 + co
        const int co = idx & 63;
        const int ci = (idx >> 6) & 31;
        const int t  = idx >> 11;
        ldsB[(t * 64 + co) * 34 + ci] = weff[idx];
    }

    const int blk = blockIdx.x;
    const int n   = blk / 961;
    const int rem = blk % 961;
    const int zd  = rem / 31;   // do
    const int zh  = rem % 31;   // ho

    // ---- stage x tile (coalesced b128 global loads, transposed scatter to LDS) ----
    const float* xb = x + (size_t)n * 1048576 + (size_t)zd * 1024 + (size_t)zh * 32;
    #pragma unroll
    for (int j = 0; j < 4; ++j) {
        const int v  = tid + j * 256;           // vec4 index, 1024 total
        const int f  = v * 4;
        const int ci = f >> 7;
        const int z  = (f >> 6) & 1;
        const int y  = (f >> 5) & 1;
        const int xw = f & 31;                  // multiple of 4
        v4f val = *(const v4f*)(xb + (size_t)ci * 32768 + z * 1024 + y * 32 + xw);
        const int q = z * 2 + y;
        #pragma unroll
        for (int e = 0; e < 4; ++e)
            ldsX[(q * 32 + xw + e) * 34 + ci] = val[e];
    }
    __syncthreads();

    const int wave = tid >> 5;
    const int lane = tid & 31;
    const int half = lane >> 4;       // lane group (0: lanes 0-15, 1: 16-31)
    const int l16  = lane & 15;
    const int co0  = (wave & 3) << 4; // co tile
    const int wo0  = (wave >> 2) << 4;// wo tile (0 or 16)
    const int co   = co0 + l16;

    v8f acc = {0.f, 0.f, 0.f, 0.f, 0.f, 0.f, 0.f, 0.f};

    #pragma unroll
    for (int t = 0; t < 8; ++t) {
        const int a = (t >> 2) & 1, b = (t >> 1) & 1, c = t & 1;
        int col = wo0 + l16 + c;
        col = (col > 31) ? 31 : col;            // clamp (OOB column is discarded)
        const int q = a * 2 + b;
        // A: VGPR j, lane-half h -> K = j + 2h  (ci = kc*4 + j + 2h), M = l16
        const float* ap = &ldsX[(q * 32 + col) * 34 + 2 * half];
        // B: VGPR j, lane-half h -> row K = j + 2h, col N = l16
        const float* bp = &ldsB[(t * 64 + co) * 34 + 2 * half];
        #pragma unroll
        for (int kc = 0; kc < 8; ++kc) {
            v2f A = *(const v2f*)(ap + kc * 4);  // even word offset -> 8B aligned
            v2f B = *(const v2f*)(bp + kc * 4);
            acc = __builtin_amdgcn_wmma_f32_16x16x4_f32(
                false, A, false, B, (short)0, acc, false, false);
        }
    }

    // ---- epilogue: D layout: N = l16 (co), M = r + 8*half (wo) ----
    const float bb   = beta[co];
    const int mbase  = half << 3;
    const size_t obase =
        ((((size_t)n * 64 + co) * 31 + zd) * 31 + zh) * 31 + wo0 + mbase;
    if (wo0 == 0) {
        #pragma unroll
        for (int r = 0; r < 8; ++r) out[obase + r] = acc[r] + bb;
    } else {
        #pragma unroll
        for (int r = 0; r < 8; ++r)
            if (wo0 + mbase + r < 31) out[obase + r] = acc[r] + bb;
    }
}

extern "C" void kernel_launch(void* const* d_in, const int* in_sizes, int n_in,
                              void* d_out, int out_size, void* d_ws, size_t ws_size,
                              hipStream_t stream) {
    const float* x         = (const float*)d_in[0]; // (16,32,32,32,32)
    const float* weight    = (const float*)d_in[1]; // (32,64,3,3,3)
    const float* conv_bias = (const float*)d_in[2]; // (64,)
    const float* bias      = (const float*)d_in[3]; // (64,1,1,1)
    const float* s1        = (const float*)d_in[4];
    const float* s2        = (const float*)d_in[5];

    float* weff = (float*)d_ws;            // 8*32*64 floats
    float* beta = weff + 8 * 32 * 64;      // 64 floats

    ModelNew_prep_kernel<<<64, 256, 0, stream>>>(weight, conv_bias, bias, s1, s2,
                                                 weff, beta);

    const int nblocks = 16 * 31 * 31;      // (n, do, ho)
    ModelNew_fused_kernel<<<nblocks, 256, 0, stream>>>(x, weff, beta,
                                                       (float*)d_out);
}